// LightGCN_10746008175456
// MI455X (gfx1250) — compile-verified
//
#include <hip/hip_runtime.h>

#define DIM 64

typedef __attribute__((ext_vector_type(2))) float v2f;
typedef __attribute__((ext_vector_type(8))) float v8f;

// ---------------------------------------------------------------------------
// K1: buf0 = concat(emb_user, emb_item); buf1 = buf2 = 0   (float4 vectorized)
// ---------------------------------------------------------------------------
__global__ void lgcn_init(const float* __restrict__ eu, const float* __restrict__ ei,
                          float* __restrict__ b0, float* __restrict__ b1,
                          float* __restrict__ b2,
                          long long userElems, long long totalElems) {
  long long i4 = (long long)blockIdx.x * blockDim.x + threadIdx.x;
  long long f = i4 * 4;
  if (f >= totalElems) return;
  float4 v;
  if (f < userElems) v = *(const float4*)(eu + f);
  else               v = *(const float4*)(ei + (f - userElems));
  *(float4*)(b0 + f) = v;
  *(float4*)(b1 + f) = make_float4(0.f, 0.f, 0.f, 0.f);
  *(float4*)(b2 + f) = make_float4(0.f, 0.f, 0.f, 0.f);
}

// ---------------------------------------------------------------------------
// K2: COO SpMM  dst[row] += val * src[col]  — one wave32 per edge,
// each lane owns a float2 slice of the 64-dim row.
//  * edge index forced scalar (readfirstlane) -> one load set per wave
//  * edge stream read non-temporally (TH=NT): touched once, must not evict
//    the 76.8MB embedding table from the 192MB L2
//  * stream-ahead speculative prefetch (global_prefetch_b8)
//  * accumulate with native f32 agent-scope atomics (no-return -> STOREcnt)
// ---------------------------------------------------------------------------
__global__ void lgcn_spmm(const float* __restrict__ vals, const int* __restrict__ row,
                          const int* __restrict__ col, const float* __restrict__ src,
                          float* __restrict__ dst, int E) {
  int wave = blockIdx.x * (blockDim.x >> 5) + (threadIdx.x >> 5);
  int lane = threadIdx.x & 31;
  if (wave >= E) return;
  int e = __builtin_amdgcn_readfirstlane(wave);  // explicitly wave-uniform

  // speculative low-locality prefetch of the edge stream ~8K edges ahead
  __builtin_prefetch(&row[e + 8192], 0, 0);
  __builtin_prefetch(&col[e + 8192], 0, 0);
  __builtin_prefetch(&vals[e + 8192], 0, 0);

  int   r = __builtin_nontemporal_load(&row[e]);
  int   c = __builtin_nontemporal_load(&col[e]);
  float v = __builtin_nontemporal_load(&vals[e]);

  const float2 s = *(const float2*)(src + (long long)c * DIM + lane * 2);
  float* d = dst + (long long)r * DIM + lane * 2;
  __hip_atomic_fetch_add(d + 0, v * s.x, __ATOMIC_RELAXED, __HIP_MEMORY_SCOPE_AGENT);
  __hip_atomic_fetch_add(d + 1, v * s.y, __ATOMIC_RELAXED, __HIP_MEMORY_SCOPE_AGENT);
}

// ---------------------------------------------------------------------------
// K3: scores. One wave32 per 16 batch rows. U,P,N rows are (b0+b1+b2) gathered
// on the fly; diag(U*P^T) and diag(U*N^T) via V_WMMA_F32_16X16X4_F32
// accumulating 16 chunks of K=4; final scale 1/9 (layer-average applied to
// both operands). Pos/neg share the A fragment.
//
// f32 16x16x4 fragment addressing (ISA 7.12.2): A lane l holds
// [M=l&15, K = 4k + 2*(l>>4) + {0,1}]; B mirrors with N across lanes —
// identical per-lane address pattern, row index swapped for the item rows.
// C/D: VGPR v, lanes 0-15 -> (M=v, N=lane), lanes 16-31 -> (M=v+8, N=lane-16).
// ---------------------------------------------------------------------------
__global__ void lgcn_score(const float* __restrict__ b0, const float* __restrict__ b1,
                           const float* __restrict__ b2, const int* __restrict__ un,
                           const int* __restrict__ pn, const int* __restrict__ nn,
                           float* __restrict__ out, int B) {
  int wave = blockIdx.x * (blockDim.x >> 5) + (threadIdx.x >> 5);
  int lane = threadIdx.x & 31;
  if (wave * 16 >= B) return;

  int m    = lane & 15;
  int half = lane >> 4;
  long long ub = (long long)un[wave * 16 + m] * DIM;
  long long pb = (long long)pn[wave * 16 + m] * DIM;
  long long nb = (long long)nn[wave * 16 + m] * DIM;

  v8f cp = {0.f, 0.f, 0.f, 0.f, 0.f, 0.f, 0.f, 0.f};
  v8f cn = {0.f, 0.f, 0.f, 0.f, 0.f, 0.f, 0.f, 0.f};

#pragma unroll
  for (int k = 0; k < 16; ++k) {
    long long cb = 4 * k + 2 * half;
    float2 a0 = *(const float2*)(b0 + ub + cb);
    float2 a1 = *(const float2*)(b1 + ub + cb);
    float2 a2 = *(const float2*)(b2 + ub + cb);
    float2 p0 = *(const float2*)(b0 + pb + cb);
    float2 p1 = *(const float2*)(b1 + pb + cb);
    float2 p2 = *(const float2*)(b2 + pb + cb);
    float2 q0 = *(const float2*)(b0 + nb + cb);
    float2 q1 = *(const float2*)(b1 + nb + cb);
    float2 q2 = *(const float2*)(b2 + nb + cb);

    v2f A;  A.x  = a0.x + a1.x + a2.x;  A.y  = a0.y + a1.y + a2.y;
    v2f Bp; Bp.x = p0.x + p1.x + p2.x;  Bp.y = p0.y + p1.y + p2.y;
    v2f Bn; Bn.x = q0.x + q1.x + q2.x;  Bn.y = q0.y + q1.y + q2.y;

    cp = __builtin_amdgcn_wmma_f32_16x16x4_f32(false, A, false, Bp, (short)0, cp,
                                               false, false);
    cn = __builtin_amdgcn_wmma_f32_16x16x4_f32(false, A, false, Bn, (short)0, cn,
                                               false, false);
  }

  // Diagonal extraction: lanes 0-7 hold diag i = lane (VGPR lane);
  // lanes 24-31 hold diag i = lane-16 (VGPR lane-24).
  float sp = 0.f, sn = 0.f;
  int sel = (lane < 8) ? lane : (lane - 24);
#pragma unroll
  for (int i = 0; i < 8; ++i) {
    if (sel == i) { sp = cp[i]; sn = cn[i]; }
  }
  if (lane < 8 || lane >= 24) {
    int oi = wave * 16 + ((lane < 8) ? lane : (lane - 16));
    const float scale = 1.0f / 9.0f;  // (1/3)^2 layer average on both operands
    out[oi]     = sp * scale;
    out[B + oi] = sn * scale;
  }
}

// ---------------------------------------------------------------------------
extern "C" void kernel_launch(void* const* d_in, const int* in_sizes, int n_in,
                              void* d_out, int out_size, void* d_ws, size_t ws_size,
                              hipStream_t stream) {
  const float* emb_user = (const float*)d_in[0];
  const float* emb_item = (const float*)d_in[1];
  const float* vals     = (const float*)d_in[2];
  const int*   row      = (const int*)d_in[3];
  const int*   col      = (const int*)d_in[4];
  const int*   un       = (const int*)d_in[5];
  const int*   pn       = (const int*)d_in[6];
  const int*   nn       = (const int*)d_in[7];

  long long userElems  = (long long)in_sizes[0];
  long long totalElems = (long long)in_sizes[0] + (long long)in_sizes[1];
  int E = in_sizes[2];
  int B = in_sizes[5];

  float* b0 = (float*)d_ws;
  float* b1 = b0 + totalElems;
  float* b2 = b1 + totalElems;
  float* out = (float*)d_out;

  // K1: concat + zero
  long long q = totalElems / 4;
  int g1 = (int)((q + 255) / 256);
  lgcn_init<<<g1, 256, 0, stream>>>(emb_user, emb_item, b0, b1, b2,
                                    userElems, totalElems);

  // K2/K3: two propagation layers (one wave per edge, 8 waves per block)
  int g2 = (E + 7) / 8;
  lgcn_spmm<<<g2, 256, 0, stream>>>(vals, row, col, b0, b1, E);
  lgcn_spmm<<<g2, 256, 0, stream>>>(vals, row, col, b1, b2, E);

  // K4: WMMA scoring (16 batch rows per wave, 8 waves per block)
  int tiles = (B + 15) / 16;
  int g3 = (tiles + 7) / 8;
  lgcn_score<<<g3, 256, 0, stream>>>(b0, b1, b2, un, pn, nn, out, B);
}